// CoxPHLoss_61272003445461
// MI455X (gfx1250) — compile-verified
//
#include <hip/hip_runtime.h>
#include <hip/hip_bf16.h>
#include <math.h>

#define BB   64
#define NN   16384
#define II   8
#define TPB  1024
#define EPT  16        // elements per thread (TPB*EPT == NN)

static_assert(TPB * EPT == NN, "coverage");

typedef float v2f __attribute__((ext_vector_type(2)));
typedef float v8f __attribute__((ext_vector_type(8)));

typedef __attribute__((address_space(1))) int as1_int;
typedef __attribute__((address_space(3))) int as3_int;

// ---- feature detection (compile-only loop: guards keep compile green) ----
#if defined(__has_builtin)
#  if __has_builtin(__builtin_amdgcn_wmma_f32_16x16x4_f32)
#    define HAVE_WMMA_F32X4 1
#  endif
#  if __has_builtin(__builtin_amdgcn_global_load_async_to_lds_b32)
#    define HAVE_ASYNC_LDS 1
#  endif
#endif

__device__ __forceinline__ void wait_async0() {
#if defined(__has_builtin) && __has_builtin(__builtin_amdgcn_s_wait_asynccnt)
  __builtin_amdgcn_s_wait_asynccnt(0);
#else
  asm volatile("s_wait_asynccnt 0" ::: "memory");
#endif
}

// Sum a float across all 32 lanes of a wave. Uses V_WMMA_F32_16X16X4_F32:
// A (16x4 f32): VGPR0 = lane value (A[m][0]=x_m for lanes 0-15, A[m][2]=x_{m+16}
// for lanes 16-31), VGPR1 = 0. B = all ones (layout-independent). Then
// D[m][n] = x_m + x_{m+16}; each lane sums its 8 D VGPRs (rows 0-7 or 8-15)
// and one shfl_xor(16) completes the total. All math in fp32.
__device__ __forceinline__ float wave_sum32(float x) {
#if HAVE_WMMA_F32X4
  v2f a; a.x = x;    a.y = 0.0f;
  v2f b; b.x = 1.0f; b.y = 1.0f;
  v8f c = {};
  v8f d = __builtin_amdgcn_wmma_f32_16x16x4_f32(
      /*neg_a=*/false, a, /*neg_b=*/false, b,
      /*c_mod=*/(short)0, c, /*reuse_a=*/false, /*reuse_b=*/false);
  float p = d[0] + d[1] + d[2] + d[3] + d[4] + d[5] + d[6] + d[7];
  p += __shfl_xor(p, 16, 32);
  return p;
#else
  for (int off = 16; off > 0; off >>= 1) x += __shfl_xor(x, off, 32);
  return x;
#endif
}

// One workgroup per (b,i) column: sort 16K elements in LDS, scan, reduce.
__global__ void __launch_bounds__(TPB, 1)
cox_column_kernel(const float* __restrict__ logh,
                  const float* __restrict__ events,
                  const float* __restrict__ durations,
                  float* __restrict__ ws) {
  __shared__ unsigned long long keys[NN];   // 128 KB (CDNA5: 320 KB LDS/WGP)
  __shared__ float redA[32];
  __shared__ float redB[32];
  __shared__ float woff[32];
  __shared__ float sgamma;

  const int tid  = threadIdx.x;
  const int lane = tid & 31;
  const int wid  = tid >> 5;
  const int col  = blockIdx.x;           // col = b*II + i  (matches loss_flat order)
  const int b    = col >> 3;
  const int i    = col & 7;
  const size_t base_g = (size_t)b * NN * II + (size_t)i;

  unsigned* k32 = (unsigned*)keys;

  // ---- Phase A: build sort keys (dur_bits<<32 | ~n) and column max of logh ----
  float lmax = -3.4e38f;
  for (int u = 0; u < EPT; ++u) {
    int n = (u << 10) + tid;                       // strided: coalesced loads
    size_t g = base_g + (size_t)n * II;
    lmax = fmaxf(lmax, logh[g]);
#if HAVE_ASYNC_LDS
    k32[2 * n] = ~(unsigned)n;                     // low dword = ~index (tie break)
    {
      // AS1 src: pointer->integer->AS1 pointer (same 64-bit address value).
      as1_int* gsrc = (as1_int*)(unsigned long long)(durations + g);
      // AS3 dst: flat->integer->low 32 bits == LDS byte offset (flat LDS
      // aperture keeps the LDS address in addr[31:0]; addrspacecast == trunc).
      as3_int* ldst = (as3_int*)(unsigned)(unsigned long long)(&k32[2 * n + 1]);
      __builtin_amdgcn_global_load_async_to_lds_b32(gsrc, ldst, 0, 0);
    }
#else
    keys[n] = ((unsigned long long)__float_as_uint(durations[g]) << 32)
              | (unsigned)(~(unsigned)n);
#endif
  }
  for (int off = 16; off > 0; off >>= 1) lmax = fmaxf(lmax, __shfl_xor(lmax, off, 32));
  if (lane == 0) redA[wid] = lmax;
#if HAVE_ASYNC_LDS
  wait_async0();
#endif
  __syncthreads();
  if (wid == 0) {
    float m = redA[lane];
    for (int off = 16; off > 0; off >>= 1) m = fmaxf(m, __shfl_xor(m, off, 32));
    if (lane == 0) sgamma = m;
  }
  __syncthreads();
  const float gamma = sgamma;

  // ---- Phase B: bitonic sort, descending (durations desc, ties: index asc) ----
  for (unsigned k = 2; k <= (unsigned)NN; k <<= 1) {
    for (unsigned j = k >> 1; j > 0; j >>= 1) {
      for (int u = 0; u < EPT; ++u) {
        unsigned x = (unsigned)((u << 10) + tid);
        unsigned y = x ^ j;
        if (y > x) {
          unsigned long long va = keys[x];
          unsigned long long vb = keys[y];
          bool desc = ((x & k) == 0);
          if (desc ? (va < vb) : (va > vb)) { keys[x] = vb; keys[y] = va; }
        }
      }
      __syncthreads();
    }
  }

  // ---- Phase C: gather lh/e by sorted index; local sum of exp(lh-gamma) ----
  const int p0 = tid * EPT;                        // blocked: scan order == tid order
  float lsum = 0.0f;
  for (int u = 0; u < EPT; ++u) {
    int p = p0 + u;
    unsigned n = ~k32[2 * p];                      // recover original index
    size_t g = base_g + (size_t)n * II;
    float lh = logh[g];
    float e  = events[g];
    lsum += expf(lh - gamma);
    k32[2 * p]     = __float_as_uint(lh);          // overwrite own slot: (lh, e)
    k32[2 * p + 1] = __float_as_uint(e);
  }

  // ---- block exclusive scan of per-thread exp-sums (shfl + LDS) ----
  float inc = lsum;
  for (int d2 = 1; d2 < 32; d2 <<= 1) {
    float t = __shfl_up(inc, (unsigned)d2, 32);
    if (lane >= d2) inc += t;
  }
  if (lane == 31) redA[wid] = inc;
  __syncthreads();
  if (wid == 0) {
    float v = redA[lane];
    float iv = v;
    for (int d2 = 1; d2 < 32; d2 <<= 1) {
      float t = __shfl_up(iv, (unsigned)d2, 32);
      if (lane >= d2) iv += t;
    }
    woff[lane] = iv - v;                           // exclusive across waves
  }
  __syncthreads();
  float run = woff[wid] + (inc - lsum);            // prefix before my first element

  // ---- Phase D: cumulative LSE terms ----
  float lloss = 0.0f, lesum = 0.0f;
  for (int u = 0; u < EPT; ++u) {
    int p = p0 + u;
    float lh = __uint_as_float(k32[2 * p]);
    float e  = __uint_as_float(k32[2 * p + 1]);
    run += expf(lh - gamma);
    float lcs = logf(run + 1e-7f) + gamma;
    lloss += (lh - lcs) * e;
    lesum += e;
  }

  // ---- Phase E: block reduction via WMMA wave sums ----
  float wl = wave_sum32(lloss);
  float we = wave_sum32(lesum);
  __syncthreads();
  if (lane == 0) { redA[wid] = wl; redB[wid] = we; }
  __syncthreads();
  if (wid == 0) {
    float tl = wave_sum32(redA[lane]);
    float te = wave_sum32(redB[lane]);
    if (lane == 0) {
      float raw = -tl;
      ws[col] = raw / (te > 0.0f ? te : 1.0f);
    }
  }
}

// Masked mean over the 512 per-column losses. One wave; WMMA reduction.
__global__ void cox_finalize_kernel(const float* __restrict__ ws,
                                    float* __restrict__ out) {
  const int lane = threadIdx.x;                    // 32 threads
  float s = 0.0f, c = 0.0f;
  for (int k = 0; k < (BB * II) / 32; ++k) {
    float v = ws[k * 32 + lane];
    if (v > 0.0f) { s += v; c += 1.0f; }
  }
  float ts = wave_sum32(s);
  float tc = wave_sum32(c);
  if (lane == 0) out[0] = ts / fmaxf(tc, 1.0f);
}

extern "C" void kernel_launch(void* const* d_in, const int* in_sizes, int n_in,
                              void* d_out, int out_size, void* d_ws, size_t ws_size,
                              hipStream_t stream) {
  (void)in_sizes; (void)n_in; (void)out_size; (void)ws_size;
  const float* logh      = (const float*)d_in[0];
  const float* events    = (const float*)d_in[1];
  const float* durations = (const float*)d_in[2];
  float* out = (float*)d_out;
  float* ws  = (float*)d_ws;                       // 512 floats of scratch

  cox_column_kernel<<<dim3(BB * II), dim3(TPB), 0, stream>>>(logh, events, durations, ws);
  cox_finalize_kernel<<<dim3(1), dim3(32), 0, stream>>>(ws, out);
}